// CustomTradingLoss_44762149159049
// MI455X (gfx1250) — compile-verified
//
#include <hip/hip_runtime.h>
#include <hip/hip_bf16.h>

// ---------------------------------------------------------------------------
// CustomTradingLoss for MI455X (gfx1250, wave32)
// loss = 0.85*(S_wce/B)/(S_abs/B+eps) + 0.15*S_ce/B + 0.1*S_tr/B
// Kernel 1: HBM-bound streaming pass -> 4 partial sums per block (deterministic)
// Kernel 2: one wave reduces the 4x2048 partials with V_WMMA_F32_16X16X4_F32
// ---------------------------------------------------------------------------

#define NB 2048   // blocks in pass 1 == partials per accumulator
#define NT 256    // threads per block (8 waves on wave32)

typedef __attribute__((ext_vector_type(2))) float v2f;
typedef __attribute__((ext_vector_type(8))) float v8f;

__global__ void __launch_bounds__(NT)
trading_loss_partials(const float* __restrict__ pred,
                      const int*   __restrict__ tgt,
                      const float* __restrict__ pc,
                      const float* __restrict__ td,
                      float* __restrict__ ws,   // [4][NB]
                      int n) {
    float s_abs = 0.f, s_wce = 0.f, s_ce = 0.f, s_tr = 0.f;

    const int stride = gridDim.x * blockDim.x;
    for (int i = blockIdx.x * blockDim.x + threadIdx.x; i < n; i += stride) {
        // predictions row: 12 contiguous bytes -> global_load_b96, dense per wave
        const float p0 = pred[3 * i + 0];
        const float p1 = pred[3 * i + 1];
        const float p2 = pred[3 * i + 2];
        const int   t  = tgt[i];

        // log-softmax cross entropy, C = 3
        const float m   = fmaxf(p0, fmaxf(p1, p2));
        const float se  = expf(p0 - m) + expf(p1 - m) + expf(p2 - m);
        const float lse = m + logf(se);
        const float pt  = (t == 0) ? p0 : ((t == 1) ? p1 : p2);
        const float ce  = lse - pt;

        const float a   = fabsf(pc[i]);
        const float tdv = td[i];
        const bool aligned = (tdv > 0.f && t == 2) || (tdv < 0.f && t == 0);

        s_abs += a;
        s_wce += ce * a;
        s_ce  += ce;
        s_tr  += aligned ? -0.1f : 0.0f;
    }

    // block tree-reduction in LDS (deterministic)
    __shared__ float4 red[NT];
    red[threadIdx.x] = make_float4(s_abs, s_wce, s_ce, s_tr);
    __syncthreads();
    for (int off = NT / 2; off > 0; off >>= 1) {
        if (threadIdx.x < off) {
            float4 me = red[threadIdx.x];
            float4 ot = red[threadIdx.x + off];
            red[threadIdx.x] = make_float4(me.x + ot.x, me.y + ot.y,
                                           me.z + ot.z, me.w + ot.w);
        }
        __syncthreads();
    }
    if (threadIdx.x == 0) {
        float4 r = red[0];
        ws[0 * NB + blockIdx.x] = r.x;  // S_abs partial
        ws[1 * NB + blockIdx.x] = r.y;  // S_wce partial
        ws[2 * NB + blockIdx.x] = r.z;  // S_ce  partial
        ws[3 * NB + blockIdx.x] = r.w;  // S_tr  partial
    }
}

// One wave (32 lanes). Each of the 4 accumulators' NB partials is consumed as
// 32 chunks of 64 floats; each chunk is a 4x16 B-operand for
// V_WMMA_F32_16X16X4_F32 with A = all-ones. Since A is all-ones, every row of
// D equals the chunk column sums, so summing ALL D entries over all lanes and
// VGPRs yields 16 * (sum of every value fed in) -- independent of the exact
// B-operand lane striping.
__global__ void __launch_bounds__(32)
trading_loss_finalize(const float* __restrict__ ws,  // [4][NB]
                      float* __restrict__ out, int n) {
    const int lane = threadIdx.x;

    __shared__ float red[32];
    __shared__ float totals[4];

    v2f ones;
    ones.x = 1.0f;
    ones.y = 1.0f;

    for (int j = 0; j < 4; ++j) {
        v8f c = {};
        #pragma unroll
        for (int chunk = 0; chunk < NB / 64; ++chunk) {
            const int idx = j * NB + chunk * 64 + lane * 2;
            v2f b;
            b.x = ws[idx];
            b.y = ws[idx + 1];
            // D = A(ones,16x4) * B(4x16) + C   (EXEC all ones: 32 uniform lanes)
            c = __builtin_amdgcn_wmma_f32_16x16x4_f32(
                    /*neg_a=*/false, ones,
                    /*neg_b=*/false, b,
                    /*c_mod=*/(short)0, c,
                    /*reuse_a=*/false, /*reuse_b=*/false);
        }
        float s = c[0] + c[1] + c[2] + c[3] + c[4] + c[5] + c[6] + c[7];
        red[lane] = s;
        __syncthreads();
        if (lane == 0) {
            float t = 0.f;
            #pragma unroll
            for (int k = 0; k < 32; ++k) t += red[k];
            totals[j] = t * (1.0f / 16.0f);  // undo the 16-row replication
        }
        __syncthreads();
    }

    if (lane == 0) {
        const double invB     = 1.0 / (double)n;
        const double mean_abs = (double)totals[0] * invB;
        const double loss     = 0.85 * ((double)totals[1] * invB) / (mean_abs + 1e-8)
                              + 0.15 * ((double)totals[2] * invB)
                              + 0.10 * ((double)totals[3] * invB);
        out[0] = (float)loss;
    }
}

extern "C" void kernel_launch(void* const* d_in, const int* in_sizes, int n_in,
                              void* d_out, int out_size, void* d_ws, size_t ws_size,
                              hipStream_t stream) {
    const float* pred = (const float*)d_in[0];  // [B,3] f32
    const int*   tgt  = (const int*)  d_in[1];  // [B]   i32
    const float* pc   = (const float*)d_in[2];  // [B]   f32
    const float* td   = (const float*)d_in[3];  // [B]   f32
    float* out = (float*)d_out;
    float* ws  = (float*)d_ws;                  // needs 4*NB*4 = 32 KB

    const int n = in_sizes[1];                  // B

    trading_loss_partials<<<NB, NT, 0, stream>>>(pred, tgt, pc, td, ws, n);
    trading_loss_finalize<<<1, 32, 0, stream>>>(ws, out, n);
}